// GATv2Layer_68324339745116
// MI455X (gfx1250) — compile-verified
//
#include <hip/hip_runtime.h>
#include <hip/hip_bf16.h>

#define N_NODES 50000
#define N_EDGES 800000
#define DIM     128
#define HEADS   8
#define HDIM    16

typedef __attribute__((ext_vector_type(2))) float v2f;
typedef __attribute__((ext_vector_type(8))) float v8f;

// ---------------------------------------------------------------------------
// Float atomic max via ordered-int trick (works with -inf init).
// ---------------------------------------------------------------------------
__device__ __forceinline__ void atomicMaxF(float* addr, float val) {
    if (val >= 0.0f)
        atomicMax((int*)addr, __float_as_int(val));
    else
        atomicMin((unsigned int*)addr, (unsigned int)__float_as_int(val));
}

// ---------------------------------------------------------------------------
// Kernel 0: init m = -inf, denom = 0, agg = 0
// ---------------------------------------------------------------------------
__global__ void k_init(float* __restrict__ m, float* __restrict__ denom,
                       float* __restrict__ agg) {
    int t = blockIdx.x * blockDim.x + threadIdx.x;
    if (t < N_NODES * HEADS) {
        m[t]     = -__builtin_inff();
        denom[t] = 0.0f;
    }
    if (t < N_NODES * DIM) agg[t] = 0.0f;
}

// ---------------------------------------------------------------------------
// Kernel 1: h_src = x @ W_src, h_dst = x @ W_dst   (fp32 WMMA 16x16x4)
// One block = 16-row panel, 8 waves each own a 16-wide N tile.
// ---------------------------------------------------------------------------
__global__ void k_gemm_src_dst(const float* __restrict__ x,
                               const float* __restrict__ Wsrc,
                               const float* __restrict__ Wdst,
                               float* __restrict__ hsrc,
                               float* __restrict__ hdst) {
    __shared__ float ax[16 * DIM];
    const int rowBase = blockIdx.x * 16;
    for (int i = threadIdx.x; i < 16 * DIM; i += 256) {
        int r = i >> 7, c = i & 127;
        ax[i] = x[(rowBase + r) * DIM + c];
    }
    __syncthreads();

    const int wave  = threadIdx.x >> 5;
    const int lane  = threadIdx.x & 31;
    const int nBase = wave * 16;
    const int mrow  = lane & 15;          // A: M = lane (both halves)
    const int khalf = (lane >> 4) * 2;    // lanes 16-31 hold K+2 / K+3
    const int ncol  = lane & 15;          // B/D: N = lane&15

    v8f accS = {}; v8f accD = {};
    for (int kk = 0; kk < DIM; kk += 4) {
        const int k0 = kk + khalf;
        v2f a;
        a.x = ax[mrow * DIM + k0];
        a.y = ax[mrow * DIM + k0 + 1];
        v2f bs, bd;
        bs.x = Wsrc[(k0    ) * DIM + nBase + ncol];
        bs.y = Wsrc[(k0 + 1) * DIM + nBase + ncol];
        bd.x = Wdst[(k0    ) * DIM + nBase + ncol];
        bd.y = Wdst[(k0 + 1) * DIM + nBase + ncol];
        accS = __builtin_amdgcn_wmma_f32_16x16x4_f32(false, a, false, bs,
                                                     (short)0, accS, false, false);
        accD = __builtin_amdgcn_wmma_f32_16x16x4_f32(false, a, false, bd,
                                                     (short)0, accD, false, false);
    }
    // D layout: reg r -> row = r + 8*(lane>>4), col = nBase + (lane&15)
    const int rOff = 8 * (lane >> 4);
    for (int r = 0; r < 8; ++r) {
        const int row = rowBase + r + rOff;
        hsrc[row * DIM + nBase + ncol] = accS[r];
        hdst[row * DIM + nBase + ncol] = accD[r];
    }
}

// ---------------------------------------------------------------------------
// Kernel 2: per-edge GATv2 logits + segment max.  One wave32 per edge.
// Lane l covers channels 4l..4l+3; head = lane>>2.
// ---------------------------------------------------------------------------
__global__ void k_edge_logits(const float* __restrict__ hsrc,
                              const float* __restrict__ hdst,
                              const int* __restrict__ ei,
                              const float* __restrict__ attn,
                              float* __restrict__ sbuf,
                              float* __restrict__ mbuf) {
    const int warp = threadIdx.x >> 5;
    const int lane = threadIdx.x & 31;
    const int e = blockIdx.x * 8 + warp;
    if (e >= N_EDGES) return;
    const int src = ei[e];
    const int dst = ei[N_EDGES + e];

    const float4 a = ((const float4*)(hsrc + (size_t)src * DIM))[lane];
    const float4 b = ((const float4*)(hdst + (size_t)dst * DIM))[lane];
    const float4 w = ((const float4*)attn)[lane];

    float t0 = a.x + b.x, t1 = a.y + b.y, t2 = a.z + b.z, t3 = a.w + b.w;
    t0 = t0 > 0.f ? t0 : 0.2f * t0;
    t1 = t1 > 0.f ? t1 : 0.2f * t1;
    t2 = t2 > 0.f ? t2 : 0.2f * t2;
    t3 = t3 > 0.f ? t3 : 0.2f * t3;
    float p = t0 * w.x + t1 * w.y + t2 * w.z + t3 * w.w;

    // reduce over the 4 lanes of this head
    p += __shfl_xor(p, 1, 32);
    p += __shfl_xor(p, 2, 32);

    if ((lane & 3) == 0) {
        const int h = lane >> 2;
        sbuf[(size_t)e * HEADS + h] = p;
        atomicMaxF(mbuf + (size_t)dst * HEADS + h, p);
    }
}

// ---------------------------------------------------------------------------
// Kernel 3: exp(s - m[dst]) in-place + segment-sum denominator.
// One thread per (edge, head).
// ---------------------------------------------------------------------------
__global__ void k_edge_exp(const int* __restrict__ ei,
                           const float* __restrict__ mbuf,
                           float* __restrict__ sbuf,
                           float* __restrict__ denom) {
    const int t = blockIdx.x * blockDim.x + threadIdx.x;
    if (t >= N_EDGES * HEADS) return;
    const int e = t >> 3, h = t & 7;
    const int dst = ei[N_EDGES + e];
    const float ex = expf(sbuf[t] - mbuf[(size_t)dst * HEADS + h]);
    sbuf[t] = ex;
    atomicAdd(denom + (size_t)dst * HEADS + h, ex);
}

// ---------------------------------------------------------------------------
// Kernel 4: weighted message scatter.  One wave32 per edge; atomics land in
// the L2-resident 25.6 MB agg buffer.
// ---------------------------------------------------------------------------
__global__ void k_edge_aggregate(const float* __restrict__ hsrc,
                                 const int* __restrict__ ei,
                                 const float* __restrict__ sbuf,
                                 const float* __restrict__ denom,
                                 float* __restrict__ agg) {
    const int warp = threadIdx.x >> 5;
    const int lane = threadIdx.x & 31;
    const int e = blockIdx.x * 8 + warp;
    if (e >= N_EDGES) return;
    const int src = ei[e];
    const int dst = ei[N_EDGES + e];
    const int h = lane >> 2;

    const float ex  = sbuf[(size_t)e * HEADS + h];
    const float den = denom[(size_t)dst * HEADS + h];
    const float alpha = ex / (den + 1e-8f);

    const float4 hs = ((const float4*)(hsrc + (size_t)src * DIM))[lane];
    float* ag = agg + (size_t)dst * DIM + lane * 4;
    atomicAdd(ag + 0, hs.x * alpha);
    atomicAdd(ag + 1, hs.y * alpha);
    atomicAdd(ag + 2, hs.z * alpha);
    atomicAdd(ag + 3, hs.w * alpha);
}

// ---------------------------------------------------------------------------
// Kernel 5: out = LN(agg @ W_out + b_out + x) * gamma + beta  (fp32 WMMA)
// ---------------------------------------------------------------------------
__global__ void k_gemm_out_ln(const float* __restrict__ agg,
                              const float* __restrict__ Wout,
                              const float* __restrict__ bout,
                              const float* __restrict__ x,
                              const float* __restrict__ gamma,
                              const float* __restrict__ beta,
                              float* __restrict__ out) {
    __shared__ float aTile[16 * DIM];
    __shared__ float oTile[16 * DIM];
    const int rowBase = blockIdx.x * 16;
    for (int i = threadIdx.x; i < 16 * DIM; i += 256) {
        int r = i >> 7, c = i & 127;
        aTile[i] = agg[(rowBase + r) * DIM + c];
    }
    __syncthreads();

    const int wave  = threadIdx.x >> 5;
    const int lane  = threadIdx.x & 31;
    const int nBase = wave * 16;
    const int mrow  = lane & 15;
    const int khalf = (lane >> 4) * 2;
    const int ncol  = lane & 15;

    v8f acc = {};
    for (int kk = 0; kk < DIM; kk += 4) {
        const int k0 = kk + khalf;
        v2f a;
        a.x = aTile[mrow * DIM + k0];
        a.y = aTile[mrow * DIM + k0 + 1];
        v2f b;
        b.x = Wout[(k0    ) * DIM + nBase + ncol];
        b.y = Wout[(k0 + 1) * DIM + nBase + ncol];
        acc = __builtin_amdgcn_wmma_f32_16x16x4_f32(false, a, false, b,
                                                    (short)0, acc, false, false);
    }
    const int rOff = 8 * (lane >> 4);
    for (int r = 0; r < 8; ++r)
        oTile[(r + rOff) * DIM + nBase + ncol] = acc[r];
    __syncthreads();

    // LayerNorm: 16 lanes per row, 8 channels per lane, shfl_xor reduction.
    const int row  = threadIdx.x >> 4;   // 0..15
    const int j    = threadIdx.x & 15;   // 0..15
    const int gRow = rowBase + row;
    float vals[8];
    float s = 0.f, ss = 0.f;
    for (int k2 = 0; k2 < 8; ++k2) {
        const int c = j * 8 + k2;
        const float v = oTile[row * DIM + c] + bout[c] + x[(size_t)gRow * DIM + c];
        vals[k2] = v; s += v; ss += v * v;
    }
    for (int msk = 1; msk < 16; msk <<= 1) {
        s  += __shfl_xor(s,  msk, 32);
        ss += __shfl_xor(ss, msk, 32);
    }
    const float mu  = s * (1.0f / DIM);
    const float var = ss * (1.0f / DIM) - mu * mu;
    const float inv = rsqrtf(var + 1e-5f);
    for (int k2 = 0; k2 < 8; ++k2) {
        const int c = j * 8 + k2;
        out[(size_t)gRow * DIM + c] = (vals[k2] - mu) * inv * gamma[c] + beta[c];
    }
}

// ---------------------------------------------------------------------------
// Launcher
// ---------------------------------------------------------------------------
extern "C" void kernel_launch(void* const* d_in, const int* in_sizes, int n_in,
                              void* d_out, int out_size, void* d_ws, size_t ws_size,
                              hipStream_t stream) {
    const float* x     = (const float*)d_in[0];
    const int*   ei    = (const int*)  d_in[1];
    const float* Wsrc  = (const float*)d_in[2];
    const float* Wdst  = (const float*)d_in[3];
    const float* attn  = (const float*)d_in[4];
    const float* Wout  = (const float*)d_in[5];
    const float* bout  = (const float*)d_in[6];
    const float* gamma = (const float*)d_in[7];
    const float* beta  = (const float*)d_in[8];
    float* out = (float*)d_out;

    // workspace layout (floats)
    float* ws    = (float*)d_ws;
    float* hsrc  = ws;                                   // N*128
    float* hdst  = hsrc  + (size_t)N_NODES * DIM;        // N*128
    float* sbuf  = hdst  + (size_t)N_NODES * DIM;        // E*8
    float* mbuf  = sbuf  + (size_t)N_EDGES * HEADS;      // N*8
    float* denom = mbuf  + (size_t)N_NODES * HEADS;      // N*8
    float* agg   = denom + (size_t)N_NODES * HEADS;      // N*128

    const int rowTiles = N_NODES / 16;          // 3125
    const int edgeBlks = N_EDGES / 8;           // 100000 (wave per edge)
    const int initBlks = (N_NODES * DIM + 255) / 256;
    const int ehBlks   = (N_EDGES * HEADS + 255) / 256;

    k_init<<<initBlks, 256, 0, stream>>>(mbuf, denom, agg);
    k_gemm_src_dst<<<rowTiles, 256, 0, stream>>>(x, Wsrc, Wdst, hsrc, hdst);
    k_edge_logits<<<edgeBlks, 256, 0, stream>>>(hsrc, hdst, ei, attn, sbuf, mbuf);
    k_edge_exp<<<ehBlks, 256, 0, stream>>>(ei, mbuf, sbuf, denom);
    k_edge_aggregate<<<edgeBlks, 256, 0, stream>>>(hsrc, ei, sbuf, denom, agg);
    k_gemm_out_ln<<<rowTiles, 256, 0, stream>>>(agg, Wout, bout, x, gamma, beta, out);
}